// Ridge_22308060136227
// MI455X (gfx1250) — compile-verified
//
#include <hip/hip_runtime.h>
#include <hip/hip_bf16.h>
#include <math.h>

// Chunked (Woodbury) recursive least squares for online ridge regression.
//   B=32 batches, N=256 steps, D=128 features, chunk T=16.
// One workgroup (8 wave32) per batch; P = (G + I)^-1 lives in LDS (~100KB,
// possible thanks to CDNA5's 320KB/WGP). The three GEMMs per chunk
// (U = X*P, S = U*X^T, P -= V^T V) run on V_WMMA_F32_16X16X4_F32 (f32 WMMA —
// the inverse recursion needs f32). Key ISA-level tricks:
//   * P is symmetric -> B fragments read row-wise -> ds_load_b64 pairs
//   * V stored transposed and pre-negated -> P update is a pure WMMA
//     accumulate with the P tile as the C operand (no load/sub/store)
//   * K-loop double buffering so DS latency overlaps the matrix pipe.

typedef __attribute__((ext_vector_type(2))) float v2f;
typedef __attribute__((ext_vector_type(8))) float v8f;

#define D     128
#define NSTEP 256
#define T     16
#define NCH   (NSTEP / T)
#define PST   130   // padded, EVEN row stride for P: keeps b64 fragments aligned

// LDS layout (floats) — all bases multiples of 4 floats (16B) for float4 use
#define OFF_P   0                      // [D][PST] inverse-Gram P (symmetric)
#define OFF_X   (OFF_P + D * PST)      // [T][D]   chunk features
#define OFF_U   (OFF_X + T * D)        // [T][D]   U = X * P
#define OFF_VN  (OFF_U + T * D)        // [D][T]   -V^T  (A operand of update)
#define OFF_VP  (OFF_VN + D * T)       // [D][T]   +V^T  (B operand of update)
#define OFF_W   (OFF_VP + D * T)       // [D]      weights
#define OFF_L   (OFF_W + D)            // [T][T]   Cholesky factor
#define OFF_S   (OFF_L + T * T)        // [T][T]   I + X P X^T
#define OFF_Y   (OFF_S + T * T)        // [T]
#define OFF_Q   (OFF_Y + T)            // [T] q = X w0
#define OFF_R   (OFF_Q + T)            // [T] r = y - q
#define OFF_Z   (OFF_R + T)            // [T] z = L^-1 r
#define OFF_A   (OFF_Z + T)            // [T] alpha = L^-T z
#define OFF_H   (OFF_A + T)            // [T] in-chunk predictions
#define SMEM_FLOATS (OFF_H + T)        // 25568 floats = 102272 bytes

__global__ __launch_bounds__(256)
void Ridge_22308060136227_kernel(const float* __restrict__ data,
                                 const float* __restrict__ targets,
                                 float* __restrict__ out) {
    extern __shared__ float smem[];
    float* Ps  = smem + OFF_P;
    float* Xs  = smem + OFF_X;
    float* Us  = smem + OFF_U;
    float* VnT = smem + OFF_VN;
    float* VpT = smem + OFF_VP;
    float* Ws  = smem + OFF_W;
    float* Ls  = smem + OFF_L;
    float* Sv  = smem + OFF_S;
    float* Ys  = smem + OFF_Y;
    float* Qv  = smem + OFF_Q;
    float* Rv  = smem + OFF_R;
    float* Zv  = smem + OFF_Z;
    float* Al  = smem + OFF_A;
    float* Yh  = smem + OFF_H;

    const int b    = blockIdx.x;
    const int tid  = threadIdx.x;
    const int lane = tid & 31;
    const int wave = tid >> 5;      // 0..7
    const int hi   = lane >> 4;     // half-wave select (K split per ISA layout)
    const int lm   = lane & 15;

    const float* Xb = data    + (size_t)b * NSTEP * D;
    const float* Yb = targets + (size_t)b * NSTEP;
    float*       Ob = out     + (size_t)b * NSTEP;

    // P = (1/LAM) I = I,  w = 0
    for (int i = tid; i < D * D; i += 256) {
        int r = i >> 7, c = i & (D - 1);
        Ps[r * PST + c] = (r == c) ? 1.0f : 0.0f;
    }
    if (tid < D) Ws[tid] = 0.0f;
    __syncthreads();

    for (int ch = 0; ch < NCH; ++ch) {
        const int t0 = ch * T;

        // ---- stage chunk X (16x128, b128 loads) and y (16) ----------------
        {
            const float4* src = (const float4*)(Xb + (size_t)t0 * D);
            float4*       dst = (float4*)Xs;
            for (int i = tid; i < T * D / 4; i += 256) dst[i] = src[i];
        }
        if (tid < T) Ys[tid] = Yb[t0 + tid];
        if (ch + 1 < NCH) {   // emits global_prefetch_b8 for next chunk
            const float* nx = Xb + (size_t)(t0 + T) * D;
            __builtin_prefetch(nx + tid * 8, 0, 1);
        }
        __syncthreads();

        // ---- q = X w0 ; r = y - q  (float4 dots) --------------------------
        if (tid < T) {
            const float4* xr = (const float4*)&Xs[tid * D];
            const float4* wr = (const float4*)Ws;
            float acc = 0.0f;
            for (int k = 0; k < D / 4; ++k) {
                float4 xv = xr[k], wv = wr[k];
                acc += xv.x * wv.x + xv.y * wv.y + xv.z * wv.z + xv.w * wv.w;
            }
            Qv[tid] = acc;
            Rv[tid] = Ys[tid] - acc;
        }
        __syncthreads();

        // ---- U = X * P : wave w owns output columns [16w,16w+16) ----------
        // B read through P's symmetry -> adjacent pair -> ds_load_b64.
        // Double-buffered K loop: loads for k+4 fly under the WMMA at k.
        {
            const int c0 = wave * 16;
            const float* arow = &Xs[lm * D];
            const float* brow = &Ps[(c0 + lm) * PST];
            v2f a0 = *(const v2f*)&arow[2 * hi];
            v2f b0 = *(const v2f*)&brow[2 * hi];
            v8f acc = {};
#pragma unroll
            for (int k = 0; k < D; k += 4) {
                v2f a1 = a0, b1 = b0;
                if (k + 4 < D) {
                    a1 = *(const v2f*)&arow[k + 4 + 2 * hi];
                    b1 = *(const v2f*)&brow[k + 4 + 2 * hi];
                }
                acc = __builtin_amdgcn_wmma_f32_16x16x4_f32(
                    false, a0, false, b0, (short)0, acc, false, false);
                a0 = a1; b0 = b1;
            }
#pragma unroll
            for (int v = 0; v < 8; ++v)
                Us[(v + 8 * hi) * D + c0 + lm] = acc[v];
        }
        __syncthreads();

        // ---- S = U X^T + I : one 16x16 tile, WMMA on wave 0 ---------------
        if (wave == 0) {
            const float* arow = &Us[lm * D];
            const float* brow = &Xs[lm * D];
            v2f a0 = *(const v2f*)&arow[2 * hi];
            v2f b0 = *(const v2f*)&brow[2 * hi];
            v8f acc = {};
#pragma unroll
            for (int k = 0; k < D; k += 4) {
                v2f a1 = a0, b1 = b0;
                if (k + 4 < D) {
                    a1 = *(const v2f*)&arow[k + 4 + 2 * hi];
                    b1 = *(const v2f*)&brow[k + 4 + 2 * hi];
                }
                acc = __builtin_amdgcn_wmma_f32_16x16x4_f32(
                    false, a0, false, b0, (short)0, acc, false, false);
                a0 = a1; b0 = b1;
            }
#pragma unroll
            for (int v = 0; v < 8; ++v) {
                const int i = v + 8 * hi;
                Sv[i * T + lm] = acc[v] + ((i == lm) ? 1.0f : 0.0f);
            }
        }
        __syncthreads();

        // ---- tiny 16x16 space: chol, z = L^-1 r, yhat, alpha = L^-T z -----
        if (tid == 0) {
            for (int i = 0; i < T; ++i) {
                float s = Sv[i * T + i];
                for (int k = 0; k < i; ++k) { float lv = Ls[i * T + k]; s -= lv * lv; }
                float di = sqrtf(s);
                Ls[i * T + i] = di;
                float inv = 1.0f / di;
                for (int j = i + 1; j < T; ++j) {
                    float s2 = Sv[j * T + i];
                    for (int k = 0; k < i; ++k) s2 -= Ls[j * T + k] * Ls[i * T + k];
                    Ls[j * T + i] = s2 * inv;
                }
            }
            for (int i = 0; i < T; ++i) {          // fwd-sub + predictions
                float s = Rv[i], p = Qv[i];
                for (int j = 0; j < i; ++j) {
                    s -= Ls[i * T + j] * Zv[j];
                    p += Ls[i * T + j] * Zv[j];    // yhat_i = q_i + L_{i,<i}.z
                }
                Yh[i] = p;
                Zv[i] = s / Ls[i * T + i];
            }
            for (int i = T - 1; i >= 0; --i) {     // back-sub
                float s = Zv[i];
                for (int j = i + 1; j < T; ++j) s -= Ls[j * T + i] * Al[j];
                Al[i] = s / Ls[i * T + i];
            }
        }
        __syncthreads();

        // ---- w += U^T alpha ; V = L^-1 U, stored as -V^T and +V^T ---------
        if (tid < D) {
            float wv = Ws[tid];
#pragma unroll
            for (int i = 0; i < T; ++i) wv += Us[i * D + tid] * Al[i];
            Ws[tid] = wv;

            float col[T];
#pragma unroll
            for (int i = 0; i < T; ++i) {
                float s = Us[i * D + tid];
                for (int j = 0; j < i; ++j) s -= Ls[i * T + j] * col[j];
                col[i] = s / Ls[i * T + i];
            }
#pragma unroll
            for (int i = 0; i < T; i += 2) {      // b64 stores
                v2f p; p[0] = col[i];  p[1] = col[i + 1];
                v2f n; n[0] = -col[i]; n[1] = -col[i + 1];
                *(v2f*)&VpT[tid * T + i] = p;
                *(v2f*)&VnT[tid * T + i] = n;
            }
        }
        if (tid < T) {
            const int g = t0 + tid;
            Ob[g] = (g == 0) ? 0.0f : Yh[tid];
        }
        __syncthreads();

        // ---- P += (-V)^T (V) : pure WMMA accumulate, C = P tile -----------
        {
            const int r0 = wave * 16;
            v2f an[4];
#pragma unroll
            for (int kk = 0; kk < 4; ++kk)        // A fragments hoisted (b64)
                an[kk] = *(const v2f*)&VnT[(r0 + lm) * T + 4 * kk + 2 * hi];
#pragma unroll
            for (int ct = 0; ct < 8; ++ct) {
                const int c0 = ct * 16;
                v2f bp[4];
#pragma unroll
                for (int kk = 0; kk < 4; ++kk)    // B fragments (b64)
                    bp[kk] = *(const v2f*)&VpT[(c0 + lm) * T + 4 * kk + 2 * hi];
                v8f acc;
#pragma unroll
                for (int v = 0; v < 8; ++v)       // C = current P tile
                    acc[v] = Ps[(r0 + v + 8 * hi) * PST + c0 + lm];
#pragma unroll
                for (int kk = 0; kk < 4; ++kk)
                    acc = __builtin_amdgcn_wmma_f32_16x16x4_f32(
                        false, an[kk], false, bp[kk], (short)0, acc, false, false);
#pragma unroll
                for (int v = 0; v < 8; ++v)
                    Ps[(r0 + v + 8 * hi) * PST + c0 + lm] = acc[v];
            }
        }
        __syncthreads();
    }
}

extern "C" void kernel_launch(void* const* d_in, const int* in_sizes, int n_in,
                              void* d_out, int out_size, void* d_ws, size_t ws_size,
                              hipStream_t stream) {
    const float* data    = (const float*)d_in[0];   // (32, 256, 128) f32
    const float* targets = (const float*)d_in[1];   // (32, 256)      f32
    float*       out     = (float*)d_out;           // (32, 256)      f32
    (void)in_sizes; (void)n_in; (void)out_size; (void)d_ws; (void)ws_size;

    constexpr size_t smem_bytes = (size_t)SMEM_FLOATS * sizeof(float); // ~100KB (CDNA5: 320KB/WGP)
    Ridge_22308060136227_kernel<<<dim3(32), dim3(256), smem_bytes, stream>>>(data, targets, out);
}